// ZoneE_77163382440506
// MI455X (gfx1250) — compile-verified
//
#include <hip/hip_runtime.h>
#include <hip/hip_bf16.h>
#include <math.h>

// ---------------- problem constants (match reference) ----------------
#define DIM_D   1024
#define DIM_DO  256
#define BATCH   4
#define SEQ_L   4096
#define NLAYERS 3
#define MTOP    (SEQ_L / 4)          // 1024
#define BL      (BATCH * SEQ_L)      // 16384 rows
#define NCHUNK  32
#define CHLEN   (SEQ_L / NCHUNK)     // 128
#define EPS_RMS  1e-6f
#define EPS_NORM 1e-12f

#define BK      16                   // K-depth per LDS stage (4 wmma steps)
#define LDSTR   20                   // padded LDS row stride (floats): 16B-aligned rows, conflict-free b64 reads

typedef __attribute__((ext_vector_type(2))) float v2f;
typedef __attribute__((ext_vector_type(8))) float v8f;

#if defined(__has_builtin)
#if __has_builtin(__builtin_amdgcn_wmma_f32_16x16x4_f32)
#define HAVE_WMMA_F32X4 1
#endif
#endif

// Device pass for gfx1250: use the CDNA5 async tensor-path via inline asm
// (portable across toolchains regardless of clang builtin signatures).
#if defined(__gfx1250__)
#define USE_ASYNC_ASM 1
#endif

// D = A(16x4,f32) * B(4x16,f32) + C(16x16,f32) on one wave32.
__device__ __forceinline__ v8f wmma_f32x4(v2f a, v2f b, v8f c) {
#ifdef HAVE_WMMA_F32X4
  return __builtin_amdgcn_wmma_f32_16x16x4_f32(false, a, false, b, (short)0, c,
                                               false, false);
#else
  c[0] += a[0] * b[0] + a[1] * b[1];  // host-pass / fallback placeholder
  return c;
#endif
}

// 16-byte global -> LDS copy via GLOBAL_LOAD_ASYNC_TO_LDS_B128 (ASYNCcnt).
// VDST = LDS byte offset (low 32 bits of a generic pointer to __shared__);
// VADDR = 64-bit global address; hardware adds LDS_BASE.
__device__ __forceinline__ void copy16_g2l(float* lds, const float* g) {
#ifdef USE_ASYNC_ASM
  unsigned loff = (unsigned)(size_t)lds;
  asm volatile("global_load_async_to_lds_b128 %0, %1, off"
               :
               : "v"(loff), "v"(g)
               : "memory");
#else
  *(float4*)lds = *(const float4*)g;
#endif
}

__device__ __forceinline__ void wait_async_le4() {
#ifdef USE_ASYNC_ASM
#if defined(__has_builtin) && __has_builtin(__builtin_amdgcn_s_wait_asynccnt)
  __builtin_amdgcn_s_wait_asynccnt(4);
#else
  asm volatile("s_wait_asynccnt 0x4" ::: "memory");
#endif
#endif
}
__device__ __forceinline__ void wait_async_le6() {
#ifdef USE_ASYNC_ASM
#if defined(__has_builtin) && __has_builtin(__builtin_amdgcn_s_wait_asynccnt)
  __builtin_amdgcn_s_wait_asynccnt(6);
#else
  asm volatile("s_wait_asynccnt 0x6" ::: "memory");
#endif
#endif
}
__device__ __forceinline__ void wait_async_all() {
#ifdef USE_ASYNC_ASM
#if defined(__has_builtin) && __has_builtin(__builtin_amdgcn_s_wait_asynccnt)
  __builtin_amdgcn_s_wait_asynccnt(0);
#else
  asm volatile("s_wait_asynccnt 0x0" ::: "memory");
#endif
#endif
}

// Lane mapping (wave32), per ISA 7.12.2:
//  A 16x4:  M = lane%16, K = 2*(lane/16) + v        (2 VGPRs)
//  B 4x16:  N = lane%16, K = 2*(lane/16) + v        (2 VGPRs)
//  C 16x16: M = e + 8*(lane/16), N = lane%16        (8 VGPRs)

// 4 k-steps (BK=16) of 16x64 per wave from staged LDS tiles.
__device__ __forceinline__ void tile_mma(const float* __restrict__ as,
                                         const float* __restrict__ ws,
                                         v8f acc[4]) {
#pragma unroll
  for (int kk = 0; kk < BK; kk += 4) {
    v2f a;
    a[0] = as[kk];
    a[1] = as[kk + 1];
#pragma unroll
    for (int j = 0; j < 4; ++j) {
      const float* wp = ws + j * 16 * LDSTR + kk;
      v2f b;
      b[0] = wp[0];
      b[1] = wp[1];
      acc[j] = wmma_f32x4(a, b, acc[j]);
    }
  }
}

// ---------------- generic C = A @ W^T  (W stored [N,K]) ----------------
// block = 128 threads = 4 waves; 64x64 output tile per block.
// Double-buffered LDS staging: async-copy tile kt+1 while WMMAs run on kt.
__global__ __launch_bounds__(128) void k_gemm(const float* __restrict__ A,
                                              const float* __restrict__ W,
                                              float* __restrict__ C,
                                              int M, int N, int K) {
  __shared__ float As[2][64 * LDSTR];
  __shared__ float Ws[2][64 * LDSTR];
  const int tid = threadIdx.x;
  const int lane = tid & 31;
  const int wave = tid >> 5;
  const int lo = lane & 15;
  const int hi = lane >> 4;
  const int mblk = blockIdx.x * 64;
  const int nblk = blockIdx.y * 64;
  const int KT = K / BK;

  // 64x16 tile = 256 float4 chunks; 128 threads -> 2 chunks each per matrix.
  auto issue = [&](int buf, int kt) {
    const int k0 = kt * BK;
#pragma unroll
    for (int i = 0; i < 2; ++i) {
      int f = tid + 128 * i;
      int row = f >> 2;
      int cq = (f & 3) * 4;
      copy16_g2l(&As[buf][row * LDSTR + cq], A + (size_t)(mblk + row) * K + k0 + cq);
      copy16_g2l(&Ws[buf][row * LDSTR + cq], W + (size_t)(nblk + row) * K + k0 + cq);
    }
  };

  v8f acc[4] = {};
  issue(0, 0);
  for (int kt = 0; kt < KT - 1; ++kt) {
    const int buf = kt & 1;
    issue(buf ^ 1, kt + 1);        // overlap next tile's global traffic
    wait_async_le4();              // tile kt complete (ASYNCcnt is in-order)
    __syncthreads();
    tile_mma(&As[buf][(wave * 16 + lo) * LDSTR + 2 * hi],
             &Ws[buf][lo * LDSTR + 2 * hi], acc);
    __syncthreads();               // everyone done reading buf before reuse
  }
  {
    const int buf = (KT - 1) & 1;
    wait_async_all();
    __syncthreads();
    tile_mma(&As[buf][(wave * 16 + lo) * LDSTR + 2 * hi],
             &Ws[buf][lo * LDSTR + 2 * hi], acc);
  }

#pragma unroll
  for (int j = 0; j < 4; ++j)
#pragma unroll
    for (int e = 0; e < 8; ++e) {
      int m = mblk + wave * 16 + e + 8 * hi;
      int n = nblk + j * 16 + lo;
      C[(size_t)m * N + n] = acc[j][e];
    }
}

// ---------------- fused gate kernel ----------------
// r = sigmoid(xc@Wr^T + br); i = sigmoid(xc@Wi^T + bi)
// a_t = exp(8*r*log_sigmoid(log_a)); gi = sqrt(max(1-a^2,0)) * i * xc
// One A fragment feeds 8 WMMAs per k-step (Wr + Wi tiles both in LDS).
__global__ __launch_bounds__(128) void k_gates(const float* __restrict__ XC,
                                               const float* __restrict__ Wr,
                                               const float* __restrict__ Wrb,
                                               const float* __restrict__ Wi,
                                               const float* __restrict__ Wib,
                                               const float* __restrict__ LA,
                                               float* __restrict__ AT,
                                               float* __restrict__ GI) {
  __shared__ float As[2][64 * LDSTR];
  __shared__ float Rs[2][64 * LDSTR];
  __shared__ float Is[2][64 * LDSTR];
  const int K = DIM_DO;
  const int tid = threadIdx.x;
  const int lane = tid & 31;
  const int wave = tid >> 5;
  const int lo = lane & 15;
  const int hi = lane >> 4;
  const int mblk = blockIdx.x * 64;
  const int nblk = blockIdx.y * 64;
  const int KT = K / BK;

  auto issue = [&](int buf, int kt) {
    const int k0 = kt * BK;
#pragma unroll
    for (int i = 0; i < 2; ++i) {
      int f = tid + 128 * i;
      int row = f >> 2;
      int cq = (f & 3) * 4;
      size_t go = (size_t)row * K + k0 + cq;
      copy16_g2l(&As[buf][row * LDSTR + cq], XC + (size_t)mblk * K + go);
      copy16_g2l(&Rs[buf][row * LDSTR + cq], Wr + (size_t)nblk * K + go);
      copy16_g2l(&Is[buf][row * LDSTR + cq], Wi + (size_t)nblk * K + go);
    }
  };

  auto mma2 = [&](int buf, v8f ar[4], v8f ai[4]) {
    const float* as = &As[buf][(wave * 16 + lo) * LDSTR + 2 * hi];
    const float* rs = &Rs[buf][lo * LDSTR + 2 * hi];
    const float* is = &Is[buf][lo * LDSTR + 2 * hi];
#pragma unroll
    for (int kk = 0; kk < BK; kk += 4) {
      v2f a;
      a[0] = as[kk];
      a[1] = as[kk + 1];
#pragma unroll
      for (int j = 0; j < 4; ++j) {
        const float* rp = rs + j * 16 * LDSTR + kk;
        const float* ip = is + j * 16 * LDSTR + kk;
        v2f br, bi;
        br[0] = rp[0]; br[1] = rp[1];
        bi[0] = ip[0]; bi[1] = ip[1];
        ar[j] = wmma_f32x4(a, br, ar[j]);
        ai[j] = wmma_f32x4(a, bi, ai[j]);
      }
    }
  };

  v8f ar[4] = {};
  v8f ai[4] = {};
  issue(0, 0);
  for (int kt = 0; kt < KT - 1; ++kt) {
    const int buf = kt & 1;
    issue(buf ^ 1, kt + 1);
    wait_async_le6();
    __syncthreads();
    mma2(buf, ar, ai);
    __syncthreads();
  }
  {
    wait_async_all();
    __syncthreads();
    mma2((KT - 1) & 1, ar, ai);
  }

#pragma unroll
  for (int j = 0; j < 4; ++j)
#pragma unroll
    for (int e = 0; e < 8; ++e) {
      int m = mblk + wave * 16 + e + 8 * hi;
      int n = nblk + j * 16 + lo;
      float r = 1.f / (1.f + expf(-(ar[j][e] + Wrb[n])));
      float ig = 1.f / (1.f + expf(-(ai[j][e] + Wib[n])));
      float ls = -log1pf(expf(-LA[n]));          // log_sigmoid(log_a)
      float av = expf(8.f * r * ls);
      float xv = XC[(size_t)m * DIM_DO + n];
      float gate = sqrtf(fmaxf(1.f - av * av, 0.f));
      AT[(size_t)m * DIM_DO + n] = av;
      GI[(size_t)m * DIM_DO + n] = gate * (ig * xv);
    }
}

// ---------------- depthwise causal conv1d (k=4) ----------------
__global__ void k_conv(const float* __restrict__ X, const float* __restrict__ CW,
                       const float* __restrict__ CB, float* __restrict__ XC) {
  int idx = blockIdx.x * blockDim.x + threadIdx.x;
  if (idx >= BATCH * SEQ_L * DIM_DO) return;
  int c = idx % DIM_DO;
  int t = (idx / DIM_DO) % SEQ_L;
  int b = idx / (DIM_DO * SEQ_L);
  const float* w = CW + c * 4;
  float acc = CB[c];
#pragma unroll
  for (int k = 0; k < 4; ++k) {
    int tt = t - 3 + k;
    if (tt >= 0) acc += X[((size_t)b * SEQ_L + tt) * DIM_DO + c] * w[k];
  }
  XC[idx] = acc;
}

// ---------------- chunked parallel linear scan: h = a*h + g ----------------
__global__ void k_scan_local(const float* __restrict__ AT, float* __restrict__ HS,
                             float* __restrict__ SP, float* __restrict__ SH) {
  int gid = blockIdx.x * blockDim.x + threadIdx.x;
  if (gid >= BATCH * NCHUNK * DIM_DO) return;
  int c = gid % DIM_DO;
  int ch = (gid / DIM_DO) % NCHUNK;
  int b = gid / (DIM_DO * NCHUNK);
  size_t base = ((size_t)b * SEQ_L + (size_t)ch * CHLEN) * DIM_DO + c;
  float h = 0.f, p = 1.f;
  for (int t = 0; t < CHLEN; ++t) {
    size_t o = base + (size_t)t * DIM_DO;
    float a = AT[o];
    float g = HS[o];         // HS aliases GI from k_gates (in-place)
    h = fmaf(a, h, g);
    p *= a;
    HS[o] = h;
  }
  size_t si = ((size_t)b * DIM_DO + c) * NCHUNK + ch;
  SP[si] = p;
  SH[si] = h;
}

__global__ void k_scan_carry(const float* __restrict__ SP,
                             const float* __restrict__ SH,
                             float* __restrict__ CR) {
  int gid = blockIdx.x * blockDim.x + threadIdx.x;
  if (gid >= BATCH * DIM_DO) return;
  size_t base = (size_t)gid * NCHUNK;
  float c = 0.f;
  for (int ch = 0; ch < NCHUNK; ++ch) {
    CR[base + ch] = c;
    c = fmaf(SP[base + ch], c, SH[base + ch]);
  }
}

__global__ void k_scan_fix(const float* __restrict__ AT, float* __restrict__ HS,
                           const float* __restrict__ CR) {
  int gid = blockIdx.x * blockDim.x + threadIdx.x;
  if (gid >= BATCH * NCHUNK * DIM_DO) return;
  int cc = gid % DIM_DO;
  int ch = (gid / DIM_DO) % NCHUNK;
  int b = gid / (DIM_DO * NCHUNK);
  float carry = CR[((size_t)b * DIM_DO + cc) * NCHUNK + ch];
  size_t base = ((size_t)b * SEQ_L + (size_t)ch * CHLEN) * DIM_DO + cc;
  float p = 1.f;
  for (int t = 0; t < CHLEN; ++t) {
    size_t o = base + (size_t)t * DIM_DO;
    p *= AT[o];
    HS[o] = fmaf(p, carry, HS[o]);
  }
}

// ---------------- RMS norm over d_outer=256 (one block per row) ----------------
__global__ __launch_bounds__(256) void k_rmsnorm(const float* __restrict__ Y,
                                                 const float* __restrict__ NW,
                                                 float* __restrict__ X) {
  __shared__ float red[256];
  int m = blockIdx.x;
  int c = threadIdx.x;
  float v = Y[(size_t)m * DIM_DO + c];
  red[c] = v * v;
  __syncthreads();
  for (int s = 128; s > 0; s >>= 1) {
    if (c < s) red[c] += red[c + s];
    __syncthreads();
  }
  float inv = rsqrtf(red[0] / (float)DIM_DO + EPS_RMS);
  X[(size_t)m * DIM_DO + c] = v * inv * NW[c];
}

// ---------------- row L2 norm over d=1024 ----------------
__global__ __launch_bounds__(256) void k_rownorm(const float* __restrict__ E,
                                                 float* __restrict__ NRM) {
  __shared__ float red[256];
  int m = blockIdx.x;
  int tid = threadIdx.x;
  const float* row = E + (size_t)m * DIM_D;
  float s = 0.f;
  for (int k = tid; k < DIM_D; k += 256) s += row[k] * row[k];
  red[tid] = s;
  __syncthreads();
  for (int st = 128; st > 0; st >>= 1) {
    if (tid < st) red[tid] += red[tid + st];
    __syncthreads();
  }
  if (tid == 0) NRM[m] = sqrtf(red[0]);
}

// ---------------- neighbor cosine -> boundary probs ----------------
__global__ __launch_bounds__(256) void k_sim(const float* __restrict__ E,
                                             const float* __restrict__ NRM,
                                             float* __restrict__ P1,
                                             float* __restrict__ P2) {
  __shared__ float red[256];
  int m = blockIdx.x;
  int tid = threadIdx.x;
  int t = m % SEQ_L;
  if (t == 0) {
    if (tid == 0) { P1[m] = 1.f; P2[m] = 1.f; }
    return;
  }
  const float* r0 = E + (size_t)m * DIM_D;
  const float* r1 = E + (size_t)(m - 1) * DIM_D;
  float s = 0.f;
  for (int k = tid; k < DIM_D; k += 256) s += r0[k] * r1[k];
  red[tid] = s;
  __syncthreads();
  for (int st = 128; st > 0; st >>= 1) {
    if (tid < st) red[tid] += red[tid + st];
    __syncthreads();
  }
  if (tid == 0) {
    float n0 = fmaxf(NRM[m], EPS_NORM);
    float n1 = fmaxf(NRM[m - 1], EPS_NORM);
    float sim = red[0] / (n0 * n1);
    float p = (1.f - sim) * 0.5f;
    P1[m] = p;
    P2[m] = p;
  }
}

// ---------------- top-M selection by rank counting (LDS-staged) ----------------
// rank(t) = #{s: p[s] > p[t]} + #{s < t: p[s] == p[t]}   (jax.top_k tie order)
__global__ __launch_bounds__(256) void k_rank(const float* __restrict__ P,
                                              int* __restrict__ FLAG) {
  __shared__ float sp[SEQ_L];
  int b = blockIdx.y;
  int t = blockIdx.x * 256 + threadIdx.x;
  const float* pb = P + (size_t)b * SEQ_L;
  for (int k = threadIdx.x; k < SEQ_L; k += 256) sp[k] = pb[k];
  __syncthreads();
  float pt = sp[t];
  int rank = 0;
  for (int s = 0; s < SEQ_L; ++s) {
    float ps = sp[s];
    rank += (ps > pt) || (ps == pt && s < t);
  }
  FLAG[(size_t)b * SEQ_L + t] = (rank < MTOP) ? 1 : 0;
}

__global__ __launch_bounds__(256) void k_compact(const int* __restrict__ FLAG,
                                                 int* __restrict__ BIDX,
                                                 float* __restrict__ BIDXF) {
  __shared__ int sf[SEQ_L];
  int b = blockIdx.y;
  int t = blockIdx.x * 256 + threadIdx.x;
  const int* fb = FLAG + (size_t)b * SEQ_L;
  for (int k = threadIdx.x; k < SEQ_L; k += 256) sf[k] = fb[k];
  __syncthreads();
  if (sf[t]) {
    int pos = 0;
    for (int s = 0; s < t; ++s) pos += sf[s];
    BIDX[b * MTOP + pos] = t;
    BIDXF[b * MTOP + pos] = (float)t;
  }
}

__global__ __launch_bounds__(256) void k_gather(const float* __restrict__ E,
                                                const int* __restrict__ BIDX,
                                                float* __restrict__ CT) {
  int j = blockIdx.x;            // 0 .. B*MTOP-1
  int b = j / MTOP;
  int idx = BIDX[j];
  const float* src = E + ((size_t)b * SEQ_L + idx) * DIM_D;
  float* dst = CT + (size_t)j * DIM_D;
  for (int k = threadIdx.x; k < DIM_D; k += 256) dst[k] = src[k];
}

// ---------------- host-side orchestration ----------------
extern "C" void kernel_launch(void* const* d_in, const int* in_sizes, int n_in,
                              void* d_out, int out_size, void* d_ws, size_t ws_size,
                              hipStream_t stream) {
  const float* h      = (const float*)d_in[0];
  const float* down_w = (const float*)d_in[1];
  const float* up_w   = (const float*)d_in[2];
  const float* conv_w = (const float*)d_in[3];
  const float* conv_b = (const float*)d_in[4];
  const float* Wr_w   = (const float*)d_in[5];
  const float* Wr_b   = (const float*)d_in[6];
  const float* Wi_w   = (const float*)d_in[7];
  const float* Wi_b   = (const float*)d_in[8];
  const float* log_a  = (const float*)d_in[9];
  const float* out_w  = (const float*)d_in[10];
  const float* norm_w = (const float*)d_in[11];

  // outputs: concept_tokens, encoder_out, boundary_probs x2, boundary_idx
  float* out     = (float*)d_out;
  float* concept = out;
  float* enc     = concept + (size_t)BATCH * MTOP * DIM_D;
  float* probs1  = enc + (size_t)BATCH * SEQ_L * DIM_D;
  float* probs2  = probs1 + (size_t)BATCH * SEQ_L;
  float* bidxf   = probs2 + (size_t)BATCH * SEQ_L;

  // workspace carve-out (~68 MB)
  float* wf = (float*)d_ws;
  const size_t nx = (size_t)BL * DIM_DO;
  float* x  = wf; wf += nx;
  float* xc = wf; wf += nx;
  float* at = wf; wf += nx;
  float* gi = wf; wf += nx;   // also holds hs (in-place scan)
  float* sp = wf; wf += (size_t)BATCH * DIM_DO * NCHUNK;
  float* sh = wf; wf += (size_t)BATCH * DIM_DO * NCHUNK;
  float* cr = wf; wf += (size_t)BATCH * DIM_DO * NCHUNK;
  float* nrm = wf; wf += (size_t)BL;
  int* flag = (int*)wf; wf += (size_t)BL;
  int* bidx = (int*)wf;

  const dim3 blkG(128);
  const dim3 blk256(256);

  // 1) down projection: x = h @ down_w^T   [16384,1024]x[1024,256]
  k_gemm<<<dim3(BL / 64, DIM_DO / 64), blkG, 0, stream>>>(h, down_w, x, BL, DIM_DO, DIM_D);

  for (int i = 0; i < NLAYERS; ++i) {
    const float* cw  = conv_w + (size_t)i * DIM_DO * 4;
    const float* cb  = conv_b + (size_t)i * DIM_DO;
    const float* wr  = Wr_w + (size_t)i * DIM_DO * DIM_DO;
    const float* wrb = Wr_b + (size_t)i * DIM_DO;
    const float* wi  = Wi_w + (size_t)i * DIM_DO * DIM_DO;
    const float* wib = Wi_b + (size_t)i * DIM_DO;
    const float* la  = log_a + (size_t)i * DIM_DO;
    const float* ow  = out_w + (size_t)i * DIM_DO * DIM_DO;
    const float* nw  = norm_w + (size_t)i * DIM_DO;

    k_conv<<<dim3((BATCH * SEQ_L * DIM_DO) / 256), blk256, 0, stream>>>(x, cw, cb, xc);
    k_gates<<<dim3(BL / 64, DIM_DO / 64), blkG, 0, stream>>>(xc, wr, wrb, wi, wib, la, at, gi);
    k_scan_local<<<dim3((BATCH * NCHUNK * DIM_DO) / 256), blk256, 0, stream>>>(at, gi, sp, sh);
    k_scan_carry<<<dim3((BATCH * DIM_DO) / 256), blk256, 0, stream>>>(sp, sh, cr);
    k_scan_fix<<<dim3((BATCH * NCHUNK * DIM_DO) / 256), blk256, 0, stream>>>(at, gi, cr);
    // y = hs @ out_w^T (into xc), then rms-norm into x
    k_gemm<<<dim3(BL / 64, DIM_DO / 64), blkG, 0, stream>>>(gi, ow, xc, BL, DIM_DO, DIM_DO);
    k_rmsnorm<<<dim3(BL), blk256, 0, stream>>>(xc, nw, x);
  }

  // 3) up projection: encoder_out = x @ up_w^T   [16384,256]x[256,1024]
  k_gemm<<<dim3(BL / 64, DIM_D / 64), blkG, 0, stream>>>(x, up_w, enc, BL, DIM_D, DIM_DO);

  // 4) router
  k_rownorm<<<dim3(BL), blk256, 0, stream>>>(enc, nrm);
  k_sim<<<dim3(BL), blk256, 0, stream>>>(enc, nrm, probs1, probs2);
  k_rank<<<dim3(SEQ_L / 256, BATCH), blk256, 0, stream>>>(probs1, flag);
  k_compact<<<dim3(SEQ_L / 256, BATCH), blk256, 0, stream>>>(flag, bidx, bidxf);
  k_gather<<<dim3(BATCH * MTOP), blk256, 0, stream>>>(enc, bidx, concept);
}